// AssembleModule_82686710382598
// MI455X (gfx1250) — compile-verified
//
#include <hip/hip_runtime.h>

// ---------------------------------------------------------------------------
// AssembleModule: out[b,t,:] = select(CLS | X[b,t-1] | RING | Xr[b,t-lx-2] |
//                                     END | 0) per packed-sequence layout.
// Pure gather: 0 FLOPs, ~280 MB HBM traffic -> bandwidth-bound (~13 us floor
// at 23.3 TB/s). Strategy: one 3 KB row per 192-thread block (6 wave32),
// block-uniform source select, B128 moves. Bulk rows use the CDNA5 async
// global<->LDS B128 path (ASYNCcnt-tracked); zero/pad rows and the fallback
// path use non-temporal B128 stores.
// ---------------------------------------------------------------------------

typedef float f4 __attribute__((ext_vector_type(4)));
typedef int   v4i __attribute__((ext_vector_type(4)));

#define KB 16
#define KLX 2048
#define KLR 1024
#define KD 768
#define KT (KLX + KLR + 3)   /* 3075 */
#define KD4 (KD / 4)         /* 192 float4 per row */
#define KROWS (KB * KT)      /* 49200 rows */

#if defined(__HIP_DEVICE_COMPILE__)
#  if __has_builtin(__builtin_amdgcn_global_load_async_to_lds_b128) && \
      __has_builtin(__builtin_amdgcn_global_store_async_from_lds_b128) && \
      __has_builtin(__builtin_amdgcn_s_wait_asynccnt)
#    define USE_ASYNC_LDS 1
#  else
#    define USE_ASYNC_LDS 0
#  endif
#else
#  define USE_ASYNC_LDS 0
#endif

// Global (AS1) / LDS (AS3) typed-pointer casts for the async builtins:
// clang expects the global side as 'v4i __attribute__((address_space(1)))*'.
#define GPTR(p) ((__attribute__((address_space(1))) v4i*)(p))
#define LPTR(p) ((__attribute__((address_space(3))) v4i*)(p))

__device__ __forceinline__ const f4* row_source(
    const float* __restrict__ X, const float* __restrict__ Xr,
    const float* __restrict__ CLS, const float* __restrict__ RING,
    const float* __restrict__ END, int b, int t, int lxb, int lrb) {
  if (t == 0) return (const f4*)CLS;
  if (t <= lxb) return (const f4*)(X + ((size_t)b * KLX + (size_t)(t - 1)) * KD);
  if (t == lxb + 1) return (const f4*)RING;
  if (t < lxb + 2 + lrb)
    return (const f4*)(Xr + ((size_t)b * KLR + (size_t)(t - lxb - 2)) * KD);
  if (t == lxb + lrb + 2) return (const f4*)END;
  return nullptr;  // zero padding region
}

__global__ __launch_bounds__(KD4) void assemble_rows_kernel(
    const float* __restrict__ X, const float* __restrict__ Xr,
    const float* __restrict__ CLS, const float* __restrict__ RING,
    const float* __restrict__ END, const int* __restrict__ lx,
    const int* __restrict__ lr, float* __restrict__ out) {
  const int tid = threadIdx.x;  // 0..191, lane-groups of 32 stay in-row
#if USE_ASYNC_LDS
  __shared__ f4 stage[KD4];     // 3 KB staging tile, per-wave-disjoint slices
#endif

  for (int row = blockIdx.x; row < KROWS; row += gridDim.x) {
    const int b = row / KT;           // block-uniform -> scalar ops
    const int t = row - b * KT;
    const int lxb = lx[b];            // uniform -> s_load
    const int lrb = lr[b];

    const f4* src = row_source(X, Xr, CLS, RING, END, b, t, lxb, lrb);
    f4* dst = (f4*)out + (size_t)row * KD4 + tid;

    if (src != nullptr) {
#if USE_ASYNC_LDS
      // CDNA5 async DMA path: global -> LDS -> global, B128 per lane.
      // Each wave touches a disjoint 512 B LDS slice, so a per-wave
      // s_wait_asynccnt(0) fully orders its load->store->reuse chain.
      __builtin_amdgcn_global_load_async_to_lds_b128(
          GPTR(src + tid), LPTR(&stage[tid]), 0, 0);
      __builtin_amdgcn_s_wait_asynccnt(0);
      __builtin_amdgcn_global_store_async_from_lds_b128(
          GPTR(dst), LPTR(&stage[tid]), 0, 0);
      __builtin_amdgcn_s_wait_asynccnt(0);
#else
      // Direct B128 copy, non-temporal (stream-once data on both sides).
      f4 v = __builtin_nontemporal_load((f4*)(src + tid));
      __builtin_nontemporal_store(v, dst);
#endif
    } else {
      f4 z = (f4)0.0f;
      __builtin_nontemporal_store(z, dst);
    }
  }
}

extern "C" void kernel_launch(void* const* d_in, const int* in_sizes, int n_in,
                              void* d_out, int out_size, void* d_ws, size_t ws_size,
                              hipStream_t stream) {
  (void)in_sizes; (void)n_in; (void)out_size; (void)d_ws; (void)ws_size;
  const float* X    = (const float*)d_in[0];
  const float* Xr   = (const float*)d_in[1];
  const float* CLS  = (const float*)d_in[2];
  const float* RING = (const float*)d_in[3];
  const float* END  = (const float*)d_in[4];
  const int*   lx   = (const int*)d_in[5];
  const int*   lr   = (const int*)d_in[6];
  float* out = (float*)d_out;

  dim3 grid(KROWS);   // one 3 KB row per block
  dim3 block(KD4);    // 192 threads = 6 wave32
  assemble_rows_kernel<<<grid, block, 0, stream>>>(X, Xr, CLS, RING, END, lx, lr, out);
}